// SinkhornLoss_66614942761140
// MI455X (gfx1250) — compile-verified
//
#include <hip/hip_runtime.h>
#include <math.h>

// Sinkhorn loss (geomloss-style, p=2, blur=0.01, scaling=0.5, 20 steps)
// between x[4096x64] and y[4096x64] fp32.
//
// Design for MI455X: the 44 softmin sweeps dominate. Instead of materializing
// the 64MB cost matrix (44 x 64MB = 2.8GB traffic), we recompute each 16x16
// tile of C = 0.5(|x|^2+|y|^2) - x.y^T with V_WMMA_F32_16X16X4_F32 (full fp32
// precision; errors in C are amplified by 1/eps ~ 1e4 inside exp) and fuse it
// into an online logsumexp held in registers. x,y (2MB) stay L2/WGP$-resident,
// so the sweep is transcendental-VALU bound (~1.5G v_exp_f32) -- the best
// achievable bound for this op on this chip. The K-reduction uses two
// independent WMMA accumulator chains to break the back-to-back D->C
// dependency the scheduler was padding with s_delay_alu.

typedef float v2f __attribute__((ext_vector_type(2)));
typedef float v8f __attribute__((ext_vector_type(8)));

#define NPTS 4096
#define DIM 64
#define CSPLIT 4
#define COLS_PER_CHUNK (NPTS / CSPLIT)           // 1024
#define TILES_PER_CHUNK (COLS_PER_CHUNK / 16)    // 64
#define BASELOG (-8.317766166719343f)            // -log(4096)
#define BLURP 1e-4f                              // blur^p = 0.01^2

// workspace layout (float offsets)
#define OFF_DIAM 0
#define OFF_X2 64
#define OFF_Y2 (OFF_X2 + NPTS)
#define OFF_F (OFF_Y2 + NPTS)
#define OFF_G (OFF_F + NPTS)
#define OFF_PM (OFF_G + NPTS)
#define OFF_PS (OFF_PM + 2 * NPTS * CSPLIT)

__device__ __forceinline__ float eps_of(float diam2, int t) {
  // eps_t = max(diam^2 * 0.25^t, blur^2); large t => blur^2 (final extrapolation)
  return fmaxf(diam2 * exp2f(-2.0f * (float)t), BLURP);
}

// ---- prep: diameter^2 of joint bounding box ---------------------------------
__global__ void prep_minmax(const float* __restrict__ x,
                            const float* __restrict__ y,
                            float* __restrict__ W) {
  __shared__ float smn[256], smx[256];
  const int tid = threadIdx.x;
  const int d = tid & 63;
  const int g = tid >> 6;
  float mn = 3.4e38f, mx = -3.4e38f;
  for (int r = g; r < NPTS; r += 4) {
    float vx = x[r * DIM + d];
    float vy = y[r * DIM + d];
    mn = fminf(mn, fminf(vx, vy));
    mx = fmaxf(mx, fmaxf(vx, vy));
  }
  smn[tid] = mn;
  smx[tid] = mx;
  __syncthreads();
  if (tid < 64) {  // g==0 partial already in mn/mx
    for (int gg = 1; gg < 4; ++gg) {
      mn = fminf(mn, smn[tid + 64 * gg]);
      mx = fmaxf(mx, smx[tid + 64 * gg]);
    }
    float dd = mx - mn;
    smn[tid] = dd * dd;
  }
  __syncthreads();
  if (tid == 0) {
    float ssum = 0.f;
    for (int i = 0; i < 64; ++i) ssum += smn[i];
    W[OFF_DIAM] = ssum;
  }
}

// ---- prep: squared row norms ------------------------------------------------
__global__ void prep_sqnorm(const float* __restrict__ x,
                            const float* __restrict__ y,
                            float* __restrict__ W) {
  const int r = blockIdx.x * 256 + threadIdx.x;  // 0..8191
  const float* src = (r < NPTS) ? (x + r * DIM) : (y + (r - NPTS) * DIM);
  float ssum = 0.f;
#pragma unroll
  for (int k = 0; k < DIM; ++k) ssum += src[k] * src[k];
  W[(r < NPTS) ? (OFF_X2 + r) : (OFF_Y2 + (r - NPTS))] = ssum;
}

// ---- fused WMMA cost-recompute + online logsumexp (partial over a col chunk)
// wave -> (dir, row-tile, col-chunk). dir0: rows=x cols=y h=g (computes f~);
// dir1: rows=y cols=x h=f (computes g~). Writes per-row (max,sum) partials.
__global__ __launch_bounds__(256) void sink_softmin(
    const float* __restrict__ x, const float* __restrict__ y,
    float* __restrict__ W, int t, int use_h) {
  const int wave = (blockIdx.x << 3) + (threadIdx.x >> 5);  // 0..2047
  const int dir = wave >> 10;
  const int rem = wave & 1023;
  const int rtile = rem >> 2;
  const int chunk = rem & 3;
  const int lane = threadIdx.x & 31;
  const int li = lane & 15;
  const int hi = lane >> 4;

  const float eps = eps_of(W[OFF_DIAM], t);
  const float inv_eps = 1.0f / eps;

  const float* X = dir ? y : x;   // row matrix
  const float* Y = dir ? x : y;   // col matrix
  const float* X2 = W + (dir ? OFF_Y2 : OFF_X2);
  const float* Y2 = W + (dir ? OFF_X2 : OFF_Y2);
  const float* H = W + (dir ? OFF_F : OFF_G);

  const int rowbase = rtile * 16;

  // A fragments (16x4 f32 per WMMA): lanes 0-15 hold K={4k,4k+1},
  // lanes 16-31 hold K={4k+2,4k+3} -> contiguous float2 loads.
  v2f areg[16];
  const float* xp = X + (rowbase + li) * DIM + 2 * hi;
#pragma unroll
  for (int k = 0; k < 16; ++k) areg[k] = *(const v2f*)(xp + 4 * k);

  // accumulator layout: VGPR v of lane -> row rowbase + v + 8*hi, col = li
  float rowterm[8];
#pragma unroll
  for (int v = 0; v < 8; ++v)
    rowterm[v] = -0.5f * inv_eps * X2[rowbase + v + 8 * hi];

  float m[8], s[8];
#pragma unroll
  for (int v = 0; v < 8; ++v) {
    m[v] = -1.0e30f;
    s[v] = 0.0f;
  }

  for (int tc = 0; tc < TILES_PER_CHUNK; ++tc) {
    const int colbase = chunk * COLS_PER_CHUNK + tc * 16;
    const float* yp = Y + (colbase + li) * DIM + 2 * hi;
    // two independent accumulation chains (even/odd K-steps) so the scheduler
    // can interleave WMMAs instead of stalling on the D->C chain
    v8f acc0 = {};
    v8f acc1 = {};
#pragma unroll
    for (int k = 0; k < 16; k += 2) {
      v2f b0 = *(const v2f*)(yp + 4 * k);
      v2f b1 = *(const v2f*)(yp + 4 * (k + 1));
      acc0 = __builtin_amdgcn_wmma_f32_16x16x4_f32(false, areg[k], false, b0,
                                                   (short)0, acc0, false, false);
      acc1 = __builtin_amdgcn_wmma_f32_16x16x4_f32(false, areg[k + 1], false, b1,
                                                   (short)0, acc1, false, false);
    }
    const int col = colbase + li;
    float colterm = BASELOG - 0.5f * inv_eps * Y2[col];
    if (use_h) colterm += H[col] * inv_eps;  // uniform branch
#pragma unroll
    for (int v = 0; v < 8; ++v) {
      float a = colterm + rowterm[v] + (acc0[v] + acc1[v]) * inv_eps;
      float nm = fmaxf(m[v], a);
      s[v] = s[v] * __expf(m[v] - nm) + __expf(a - nm);
      m[v] = nm;
    }
  }

  // merge the 16 lanes holding the same 8 rows (xor masks < 16 stay in-half)
#pragma unroll
  for (int off = 1; off < 16; off <<= 1) {
#pragma unroll
    for (int v = 0; v < 8; ++v) {
      float om = __shfl_xor(m[v], off, 32);
      float os = __shfl_xor(s[v], off, 32);
      float nm = fmaxf(m[v], om);
      s[v] = s[v] * __expf(m[v] - nm) + os * __expf(om - nm);
      m[v] = nm;
    }
  }

  if (li == 0) {
    float* PM = W + OFF_PM;
    float* PS = W + OFF_PS;
#pragma unroll
    for (int v = 0; v < 8; ++v) {
      const int row = rowbase + v + 8 * hi;
      const int p = ((dir << 12) + row) * CSPLIT + chunk;
      PM[p] = m[v];
      PS[p] = s[v];
    }
  }
}

// ---- merge chunk partials; f_new = -eps*(m + log s); optional 0.5 averaging
__global__ __launch_bounds__(256) void sink_combine(float* __restrict__ W,
                                                    int t, int mode) {
  const int idx = blockIdx.x * 256 + threadIdx.x;  // 0..8191
  const int dir = idx >> 12;
  const int row = idx & (NPTS - 1);
  const float eps = eps_of(W[OFF_DIAM], t);
  const float* PM = W + OFF_PM + idx * CSPLIT;
  const float* PS = W + OFF_PS + idx * CSPLIT;
  float m = PM[0], s = PS[0];
#pragma unroll
  for (int c = 1; c < CSPLIT; ++c) {
    float om = PM[c], os = PS[c];
    float nm = fmaxf(m, om);
    s = s * __expf(m - nm) + os * __expf(om - nm);
    m = nm;
  }
  const float val = -eps * (m + __logf(s));
  float* F = W + (dir ? OFF_G : OFF_F);
  F[row] = mode ? 0.5f * (F[row] + val) : val;
}

// ---- final: mean(f) + mean(g) ----------------------------------------------
__global__ void sink_reduce(const float* __restrict__ W,
                            float* __restrict__ out) {
  __shared__ float sb[256];
  const int tid = threadIdx.x;
  float ssum = 0.f;
  for (int i = tid; i < NPTS; i += 256)
    ssum += W[OFF_F + i] + W[OFF_G + i];
  sb[tid] = ssum;
  __syncthreads();
  for (int off = 128; off > 0; off >>= 1) {
    if (tid < off) sb[tid] += sb[tid + off];
    __syncthreads();
  }
  if (tid == 0) out[0] = sb[0] * (1.0f / (float)NPTS);
}

extern "C" void kernel_launch(void* const* d_in, const int* in_sizes, int n_in,
                              void* d_out, int out_size, void* d_ws,
                              size_t ws_size, hipStream_t stream) {
  (void)in_sizes; (void)n_in; (void)out_size; (void)ws_size;
  const float* x = (const float*)d_in[0];  // synth_feature_sample
  const float* y = (const float*)d_in[1];  // input_feature_sample
  float* W = (float*)d_ws;
  float* out = (float*)d_out;

  prep_minmax<<<1, 256, 0, stream>>>(x, y, W);
  prep_sqnorm<<<(2 * NPTS) / 256, 256, 0, stream>>>(x, y, W);

  // init at eps_0 with h = b_log / a_log only
  sink_softmin<<<256, 256, 0, stream>>>(x, y, W, 0, 0);
  sink_combine<<<32, 256, 0, stream>>>(W, 0, 0);

  // 20 eps-scaling steps with 0.5 averaging
  for (int t = 0; t < 20; ++t) {
    sink_softmin<<<256, 256, 0, stream>>>(x, y, W, t, 1);
    sink_combine<<<32, 256, 0, stream>>>(W, t, 1);
  }

  // final extrapolation at eps = blur^2 (t large => clamp)
  sink_softmin<<<256, 256, 0, stream>>>(x, y, W, 1000, 1);
  sink_combine<<<32, 256, 0, stream>>>(W, 1000, 0);

  sink_reduce<<<1, 256, 0, stream>>>(W, out);
}